// GraphSAGE_67388036874504
// MI455X (gfx1250) — compile-verified
//
#include <hip/hip_runtime.h>

typedef float v2f __attribute__((ext_vector_type(2)));
typedef float v8f __attribute__((ext_vector_type(8)));

// ---------------------------------------------------------------------------
// Zero workspace (float4 stores, one-shot grid)
// ---------------------------------------------------------------------------
__global__ __launch_bounds__(256) void zero_ws_kernel(float4* __restrict__ p, long n4) {
    long i = (long)blockIdx.x * 256 + threadIdx.x;
    if (i < n4) p[i] = make_float4(0.f, 0.f, 0.f, 0.f);
}

// ---------------------------------------------------------------------------
// Degree count + reciprocal
// ---------------------------------------------------------------------------
__global__ __launch_bounds__(256) void degree_kernel(const int* __restrict__ dst,
                                                     unsigned* __restrict__ cnt, int E) {
    int e = blockIdx.x * 256 + threadIdx.x;
    if (e < E) atomicAdd(&cnt[dst[e]], 1u);
}

__global__ __launch_bounds__(256) void inv_kernel(const unsigned* __restrict__ cnt,
                                                  float* __restrict__ inv, int N) {
    int i = blockIdx.x * 256 + threadIdx.x;
    if (i < N) inv[i] = 1.0f / fmaxf((float)cnt[i], 1.0f);
}

// ---------------------------------------------------------------------------
// Edge scatter-sum, 128 features: one wave per edge, float4 per lane
// ---------------------------------------------------------------------------
__global__ __launch_bounds__(256) void scatter128_kernel(const float* __restrict__ x,
                                                         const int* __restrict__ src,
                                                         const int* __restrict__ dst,
                                                         float* __restrict__ agg, int E) {
    int e = blockIdx.x * 8 + (threadIdx.x >> 5);
    if (e >= E) return;
    int lane = threadIdx.x & 31;
    int s = src[e], d = dst[e];
    const float4 v = *(const float4*)(x + (size_t)s * 128 + lane * 4);
    float* a = agg + (size_t)d * 128 + lane * 4;
    unsafeAtomicAdd(a + 0, v.x);
    unsafeAtomicAdd(a + 1, v.y);
    unsafeAtomicAdd(a + 2, v.z);
    unsafeAtomicAdd(a + 3, v.w);
}

// ---------------------------------------------------------------------------
// Edge scatter-sum, 64 features: 16 lanes per edge, float4 per lane
// ---------------------------------------------------------------------------
__global__ __launch_bounds__(256) void scatter64_kernel(const float* __restrict__ h,
                                                        const int* __restrict__ src,
                                                        const int* __restrict__ dst,
                                                        float* __restrict__ agg, int E) {
    int t = blockIdx.x * 256 + threadIdx.x;
    int e = t >> 4;
    if (e >= E) return;
    int fl = (t & 15) * 4;
    int s = src[e], d = dst[e];
    const float4 v = *(const float4*)(h + (size_t)s * 64 + fl);
    float* a = agg + (size_t)d * 64 + fl;
    unsafeAtomicAdd(a + 0, v.x);
    unsafeAtomicAdd(a + 1, v.y);
    unsafeAtomicAdd(a + 2, v.z);
    unsafeAtomicAdd(a + 3, v.w);
}

// ---------------------------------------------------------------------------
// Fused mean + SAGE GEMM:
//   out[16x64 tile] = (agg * inv) @ W_l + bias + xin @ W_r
// One wave per 16-row tile; 4 accumulators cover all 64 output columns.
//
// Stacked weights [W_l ; W_r] are staged in LDS in K-PAIR-INTERLEAVED form:
//   w[(k/2)*128 + 2*n + (k&1)] = W_stacked[k][n]
// so each B fragment (rows k,k+1 for one column) is a single aligned
// ds_load_b64 straight into an even VGPR pair — no pack shuffles.
//
// V_WMMA_F32_16X16X4_F32 fragment layouts (ISA 7.12.2):
//   A 16x4 : lane m (0..15) | m (16..31); vgpr j holds A[m][2*(lane>>4)+j]
//   B 4x16 : lane n (0..15) | n (16..31); vgpr j holds B[2*(lane>>4)+j][n]
//   C/D    : vgpr v -> rows v (lanes 0-15) and v+8 (lanes 16-31), col = lane&15
// ---------------------------------------------------------------------------
template <int KF>
__global__ __launch_bounds__(256) void sage_gemm_wmma(const float* __restrict__ agg,
                                                      const float* __restrict__ inv,
                                                      const float* __restrict__ xin,
                                                      const float* __restrict__ Wl,
                                                      const float* __restrict__ bias,
                                                      const float* __restrict__ Wr,
                                                      float* __restrict__ out, int M) {
    extern __shared__ float w[];  // [2*KF/2 pairs][64 cols][2] interleaved
    for (int i = threadIdx.x; i < 2 * KF * 64; i += 256) {
        const int k = i >> 6;   // stacked row 0..2*KF-1
        const int n = i & 63;   // column
        const float val = (k < KF) ? Wl[i] : Wr[i - KF * 64];
        w[((k >> 1) * 64 + n) * 2 + (k & 1)] = val;
    }
    __syncthreads();

    const int lane = threadIdx.x & 31;
    const int wave = threadIdx.x >> 5;
    const int tile = blockIdx.x * 8 + wave;
    if (tile * 16 >= M) return;  // wave-uniform guard: EXEC stays all-ones for WMMA

    const int n0 = lane & 15;
    const int half = lane >> 4;
    const int row = tile * 16 + n0;
    const float s = inv[row];
    const float* arow = agg + (size_t)row * KF;
    const float* xrow = xin + (size_t)row * KF;

    v8f c0 = {}, c1 = {}, c2 = {}, c3 = {};

    // Pass 1: mean-aggregated neighbor features through W_l (scale fused in)
#pragma unroll 2
    for (int kk = 0; kk < KF / 4; ++kk) {
        const int k = kk * 4 + half * 2;
        v2f a;
        a.x = arow[k] * s;
        a.y = arow[k + 1] * s;
        const float* wp = w + (size_t)(kk * 2 + half) * 128;  // pair row, 64 col-pairs
        v2f b0 = *(const v2f*)(wp + 2 * n0);
        v2f b1 = *(const v2f*)(wp + 2 * (n0 + 16));
        v2f b2 = *(const v2f*)(wp + 2 * (n0 + 32));
        v2f b3 = *(const v2f*)(wp + 2 * (n0 + 48));
        c0 = __builtin_amdgcn_wmma_f32_16x16x4_f32(false, a, false, b0, (short)0, c0, false, false);
        c1 = __builtin_amdgcn_wmma_f32_16x16x4_f32(false, a, false, b1, (short)0, c1, false, false);
        c2 = __builtin_amdgcn_wmma_f32_16x16x4_f32(false, a, false, b2, (short)0, c2, false, false);
        c3 = __builtin_amdgcn_wmma_f32_16x16x4_f32(false, a, false, b3, (short)0, c3, false, false);
    }
    // Pass 2: self features through W_r (pair rows KF/2 .. KF-1 of stacked LDS)
#pragma unroll 2
    for (int kk = 0; kk < KF / 4; ++kk) {
        const int k = kk * 4 + half * 2;
        v2f a;
        a.x = xrow[k];
        a.y = xrow[k + 1];
        const float* wp = w + (size_t)(KF / 2 + kk * 2 + half) * 128;
        v2f b0 = *(const v2f*)(wp + 2 * n0);
        v2f b1 = *(const v2f*)(wp + 2 * (n0 + 16));
        v2f b2 = *(const v2f*)(wp + 2 * (n0 + 32));
        v2f b3 = *(const v2f*)(wp + 2 * (n0 + 48));
        c0 = __builtin_amdgcn_wmma_f32_16x16x4_f32(false, a, false, b0, (short)0, c0, false, false);
        c1 = __builtin_amdgcn_wmma_f32_16x16x4_f32(false, a, false, b1, (short)0, c1, false, false);
        c2 = __builtin_amdgcn_wmma_f32_16x16x4_f32(false, a, false, b2, (short)0, c2, false, false);
        c3 = __builtin_amdgcn_wmma_f32_16x16x4_f32(false, a, false, b3, (short)0, c3, false, false);
    }

    // Bias (per output column) + writeback per D layout
    const float bb0 = bias[n0];
    const float bb1 = bias[n0 + 16];
    const float bb2 = bias[n0 + 32];
    const float bb3 = bias[n0 + 48];
    const int rbase = tile * 16 + half * 8;
#pragma unroll
    for (int v = 0; v < 8; ++v) {
        float* orow = out + (size_t)(rbase + v) * 64;
        orow[n0] = c0[v] + bb0;
        orow[n0 + 16] = c1[v] + bb1;
        orow[n0 + 32] = c2[v] + bb2;
        orow[n0 + 48] = c3[v] + bb3;
    }
}

// ---------------------------------------------------------------------------
// Host-side orchestration
// ---------------------------------------------------------------------------
extern "C" void kernel_launch(void* const* d_in, const int* in_sizes, int n_in,
                              void* d_out, int out_size, void* d_ws, size_t ws_size,
                              hipStream_t stream) {
    const float* x   = (const float*)d_in[0];
    const int*   ei  = (const int*)d_in[1];
    const float* W1l = (const float*)d_in[2];
    const float* b1  = (const float*)d_in[3];
    const float* W1r = (const float*)d_in[4];
    const float* W2l = (const float*)d_in[5];
    const float* b2  = (const float*)d_in[6];
    const float* W2r = (const float*)d_in[7];
    float* out = (float*)d_out;

    const int N = in_sizes[0] / 128;
    const int E = in_sizes[1] / 2;
    const int* src = ei;
    const int* dst = ei + E;

    // Workspace layout (floats): agg1[N*128] | agg2[N*64] | h1[N*64] | cnt[N] | inv[N]
    float* ws = (float*)d_ws;
    float* agg1 = ws;
    float* agg2 = agg1 + (size_t)N * 128;
    float* h1 = agg2 + (size_t)N * 64;
    unsigned* cnt = (unsigned*)(h1 + (size_t)N * 64);
    float* invp = (float*)(cnt + N);

    // Zero agg1 + agg2 + h1 + cnt in one contiguous pass (N*257 floats)
    {
        long n4 = ((long)N * 257) / 4;
        long blocks = (n4 + 255) / 256;
        zero_ws_kernel<<<(unsigned)blocks, 256, 0, stream>>>((float4*)ws, n4);
    }

    // Degree + reciprocal (shared by both layers)
    degree_kernel<<<(E + 255) / 256, 256, 0, stream>>>(dst, cnt, E);
    inv_kernel<<<(N + 255) / 256, 256, 0, stream>>>(cnt, invp, N);

    // Layer 1: scatter 128-dim features, then fused mean+GEMM -> h1
    scatter128_kernel<<<(E + 7) / 8, 256, 0, stream>>>(x, src, dst, agg1, E);
    const int tiles = (N + 15) / 16;
    const int gblocks = (tiles + 7) / 8;
    sage_gemm_wmma<128><<<gblocks, 256, 2 * 128 * 64 * sizeof(float), stream>>>(
        agg1, invp, x, W1l, b1, W1r, h1, N);

    // Layer 2: scatter 64-dim features, then fused mean+GEMM -> out
    {
        long threads = (long)E * 16;
        long blocks = (threads + 255) / 256;
        scatter64_kernel<<<(unsigned)blocks, 256, 0, stream>>>(h1, src, dst, agg2, E);
    }
    sage_gemm_wmma<64><<<gblocks, 256, 2 * 64 * 64 * sizeof(float), stream>>>(
        agg2, invp, h1, W2l, b2, W2r, out, N);
}